// FlyLoRALayer_51367808860215
// MI455X (gfx1250) — compile-verified
//
#include <hip/hip_runtime.h>
#include <hip/hip_bf16.h>

// FlyLoRA eval-mode forward for MI455X (gfx1250, wave32, WMMA).
//   y   = x @ A^T                  [16384 tokens, 32]
//   m   = top-8 mask of |y + d|    (k comes from input, default 8)
//   out = (y * m) @ B^T * 2.0      [16384 tokens, 4096]
//
// Memory-bound problem (~0.51 GB moved, ~8.6 GFLOP). Both GEMMs run on
// V_WMMA_F32_16X16X4_F32 so the VALU is never the bottleneck; x is streamed
// with non-temporal loads and out with non-temporal stores, while A/B stay
// L2-resident with default temporal hints.

typedef __attribute__((ext_vector_type(2))) float v2f;
typedef __attribute__((ext_vector_type(8))) float v8f;

#define TOK_PER_BLOCK 16
#define WAVES_PER_BLOCK 4
#define THREADS_PER_BLOCK 128
#define DMODEL 4096
#define RRANK 32
#define ALPHA_OVER_R 2.0f

__device__ __forceinline__ v8f wmma_f32_16x16x4(v2f a, v2f b, v8f c) {
  return __builtin_amdgcn_wmma_f32_16x16x4_f32(
      /*neg_a=*/false, a, /*neg_b=*/false, b,
      /*c_mod=*/(short)0, c, /*reuse_a=*/false, /*reuse_b=*/false);
}

__global__ __launch_bounds__(THREADS_PER_BLOCK) void flylora_fwd_kernel(
    const float* __restrict__ x,   // [16384, 4096]
    const float* __restrict__ A,   // [32, 4096]
    const float* __restrict__ Bm,  // [4096, 32]
    const float* __restrict__ d,   // [32]
    const int* __restrict__ kptr,  // [1]
    float* __restrict__ out)       // [16384, 4096]
{
  __shared__ float sYpart[WAVES_PER_BLOCK][TOK_PER_BLOCK][RRANK]; // 8 KB
  __shared__ float sY[TOK_PER_BLOCK * RRANK];                     // 2 KB
  __shared__ float sYm[TOK_PER_BLOCK][RRANK];                     // 2 KB

  const int tid    = threadIdx.x;
  const int lane   = tid & 31;
  const int wave   = tid >> 5;           // 0..3
  const int laneLo = lane & 15;          // row/col index within 16
  const int laneHi = lane >> 4;          // 0 or 1 (K-pair / M-half select)
  const int tokBase = blockIdx.x * TOK_PER_BLOCK;

  // ---------------- Phase 1: y = x @ A^T (K split across 4 waves) ----------
  // A-operand (16x4 f32): lane holds x[tok=laneLo][k + laneHi*2 + {0,1}]
  // B-operand (4x16 f32): lane holds A[r=rBase+laneLo][k + laneHi*2 + {0,1}]
  {
    const float* xRow  = x + (size_t)(tokBase + laneLo) * DMODEL + laneHi * 2;
    const float* aRow0 = A + (size_t)(laneLo) * DMODEL + laneHi * 2;        // r 0..15
    const float* aRow1 = A + (size_t)(16 + laneLo) * DMODEL + laneHi * 2;   // r 16..31

    v8f acc0 = {0.f, 0.f, 0.f, 0.f, 0.f, 0.f, 0.f, 0.f};
    v8f acc1 = {0.f, 0.f, 0.f, 0.f, 0.f, 0.f, 0.f, 0.f};

    const int k0 = wave * (DMODEL / WAVES_PER_BLOCK);
    const int k1 = k0 + (DMODEL / WAVES_PER_BLOCK);
#pragma unroll 4
    for (int kk = k0; kk < k1; kk += 4) {
      v2f av = __builtin_nontemporal_load((const v2f*)(xRow + kk)); // x: streamed once
      v2f b0 = *(const v2f*)(aRow0 + kk);                           // A: L2-resident
      v2f b1 = *(const v2f*)(aRow1 + kk);
      acc0 = wmma_f32_16x16x4(av, b0, acc0);
      acc1 = wmma_f32_16x16x4(av, b1, acc1);
    }

    // C/D layout: VGPR i -> row M = i + laneHi*8, col = laneLo
#pragma unroll
    for (int i = 0; i < 8; ++i) {
      const int m = i + laneHi * 8;
      sYpart[wave][m][laneLo]      = acc0[i];
      sYpart[wave][m][laneLo + 16] = acc1[i];
    }
  }
  __syncthreads();

  // Cross-wave K reduction: 512 y values, 128 threads -> 4 each.
  {
    const float* p = &sYpart[0][0][0];
#pragma unroll
    for (int idx = tid; idx < TOK_PER_BLOCK * RRANK; idx += THREADS_PER_BLOCK) {
      sY[idx] = p[idx] + p[idx + 512] + p[idx + 1024] + p[idx + 1536];
    }
  }
  __syncthreads();

  // ---------------- Phase 2: per-token top-k mask on |y + d| ---------------
  if (tid < TOK_PER_BLOCK) {
    int ksel = *kptr;
    ksel = ksel < 0 ? 0 : (ksel > RRANK ? RRANK : ksel);
    float yv[RRANK], key[RRANK], ym[RRANK];
#pragma unroll
    for (int r = 0; r < RRANK; ++r) {
      yv[r]  = sY[tid * RRANK + r];
      key[r] = fabsf(yv[r] + d[r]);
      ym[r]  = 0.0f;
    }
    for (int j = 0; j < ksel; ++j) {          // iterative argmax, first-index ties
      int best = 0;
      float bk = key[0];
#pragma unroll
      for (int r = 1; r < RRANK; ++r) {
        if (key[r] > bk) { bk = key[r]; best = r; }
      }
      ym[best]  = yv[best];                   // mask applies to y, not y+d
      key[best] = -1.0f;                      // abs() >= 0, so excluded
    }
#pragma unroll
    for (int r = 0; r < RRANK; ++r) sYm[tid][r] = ym[r];
  }
  __syncthreads();

  // ---------------- Phase 3: out = (y*m) @ B^T * 2  ------------------------
  // A-operand per K-step s (K=32 -> 8 steps), hoisted once from LDS:
  //   lane holds yMasked[tok=laneLo][s*4 + laneHi*2 + {0,1}]
  v2f aOp[8];
#pragma unroll
  for (int s = 0; s < 8; ++s) {
    const int kk = s * 4 + laneHi * 2;
    v2f t = { sYm[laneLo][kk], sYm[laneLo][kk + 1] };
    aOp[s] = t;
  }

  // B-operand: lane holds B[o=oBase+laneLo][s*4 + laneHi*2 + {0,1}] (contig f2)
  const float* bLane = Bm + (size_t)laneLo * RRANK + laneHi * 2;
  for (int oBase = wave * 16; oBase < DMODEL; oBase += WAVES_PER_BLOCK * 16) {
    const float* bPtr = bLane + (size_t)oBase * RRANK;
    v8f acc = {0.f, 0.f, 0.f, 0.f, 0.f, 0.f, 0.f, 0.f};
#pragma unroll
    for (int s = 0; s < 8; ++s) {
      v2f bv = *(const v2f*)(bPtr + s * 4);   // B: L2-resident (512 KB)
      acc = wmma_f32_16x16x4(aOp[s], bv, acc);
    }
    // Scale + streaming store. Row M = i + laneHi*8, col = oBase + laneLo.
    float* oCol = out + (size_t)(tokBase + laneHi * 8) * DMODEL + oBase + laneLo;
#pragma unroll
    for (int i = 0; i < 8; ++i) {
      __builtin_nontemporal_store(ALPHA_OVER_R * acc[i],
                                  oCol + (size_t)i * DMODEL);
    }
  }
}

extern "C" void kernel_launch(void* const* d_in, const int* in_sizes, int n_in,
                              void* d_out, int out_size, void* d_ws, size_t ws_size,
                              hipStream_t stream) {
  (void)in_sizes; (void)n_in; (void)d_ws; (void)ws_size; (void)out_size;
  const float* x  = (const float*)d_in[0];  // [4,4096,4096]
  const float* A  = (const float*)d_in[1];  // [32,4096]
  const float* Bm = (const float*)d_in[2];  // [4096,32]
  const float* dv = (const float*)d_in[3];  // [32]
  const int*   kp = (const int*)d_in[4];    // [1]
  float* out = (float*)d_out;               // [4,4096,4096]

  const int tokens = 4 * 4096;
  dim3 grid(tokens / TOK_PER_BLOCK);        // 1024 blocks
  dim3 block(THREADS_PER_BLOCK);            // 4 waves (wave32)
  flylora_fwd_kernel<<<grid, block, 0, stream>>>(x, A, Bm, dv, kp, out);
}